// CoOccurWithNorm_3839700762636
// MI455X (gfx1250) — compile-verified
//
#include <hip/hip_runtime.h>
#include <stdint.h>

// Co-occurrence soft histogram, one workgroup per (b,c) image.
// CDNA5-specific paths used:
//   * 288 KB LDS per workgroup (histogram resident in LDS -- needs CDNA5's 320KB WGP LDS)
//   * global_load_async_to_lds_b128 double-buffered row staging (ASYNCcnt)
//   * s_wait_asynccnt / s_wait_dscnt split counters
//   * explicit native ds_add_f32 LDS float atomics (4-op clause per pair)
// WMMA is deliberately not used: the dense H = A^T B reformulation would do
// 128x the useful work (A/B have 2 nonzeros per 256-wide row); this scatter
// design sits at the HBM / LDS-atomic roofline instead.

#define NBINS    256
#define IMG_W    512
#define IMG_H    512
#define CH_ROWS  7                     // pair-rows processed per chunk
#define STG_ROWS 8                     // rows staged per chunk (CH_ROWS + 1)
#define NCHUNK   73                    // 73 * 7 = 511 pair-rows
#define TPB      1024                  // 32 wave32

#define HIST_ELEMS (NBINS * NBINS)     // 65536 floats = 256 KB
#define STG_ELEMS  (STG_ROWS * IMG_W)  // 4096 floats = 16 KB per buffer
#define LDS_BYTES  ((HIST_ELEMS + 2 * STG_ELEMS) * sizeof(float))  // 294912 B

__device__ __forceinline__ void soft_bins(float x, int& i0, int& i1,
                                          float& w0, float& w1) {
  float fi = fminf(fmaxf(floorf(x), 0.0f), 255.0f);
  w0 = 0.5f + 0.5f * __cosf((x - fi) * 3.14159265358979323846f);
  w1 = 1.0f - w0;
  i0 = (int)fi;
  i1 = min(i0 + 1, NBINS - 1);
}

__device__ __forceinline__ void hist_accum(uint32_t hbase, float a, float b) {
  int a0, a1, b0, b1;
  float wa0, wa1, wb0, wb1;
  soft_bins(a, a0, a1, wa0, wa1);
  soft_bins(b, b0, b1, wb0, wb1);
  uint32_t ad00 = hbase + (uint32_t)((a0 << 8) + b0) * 4u;
  uint32_t ad01 = hbase + (uint32_t)((a0 << 8) + b1) * 4u;
  uint32_t ad10 = hbase + (uint32_t)((a1 << 8) + b0) * 4u;
  uint32_t ad11 = hbase + (uint32_t)((a1 << 8) + b1) * 4u;
  // Four native LDS float atomics, issued as one back-to-back clause.
  asm volatile(
      "ds_add_f32 %0, %4\n\t"
      "ds_add_f32 %1, %5\n\t"
      "ds_add_f32 %2, %6\n\t"
      "ds_add_f32 %3, %7"
      :: "v"(ad00), "v"(ad01), "v"(ad10), "v"(ad11),
         "v"(wa0 * wb0), "v"(wa0 * wb1), "v"(wa1 * wb0), "v"(wa1 * wb1)
      : "memory");
}

__global__ __launch_bounds__(TPB) void cooccur_hist_kernel(
    const float* __restrict__ X, float* __restrict__ out) {
  extern __shared__ float smem[];
  float* hist = smem;                  // [65536]
  float* stg0 = smem + HIST_ELEMS;     // [4096]
  float* stg1 = stg0 + STG_ELEMS;      // [4096]

  const int tid = threadIdx.x;
  const int img = blockIdx.x;          // b*C + c
  const float* src = X + (size_t)img * (IMG_H * IMG_W);

  // ---- zero the LDS histogram (ds_store_b128) ----
  float4* h4 = (float4*)hist;
  const float4 z4 = make_float4(0.f, 0.f, 0.f, 0.f);
  #pragma unroll
  for (int k = 0; k < HIST_ELEMS / (4 * TPB); ++k)
    h4[k * TPB + tid] = z4;

  // Each chunk = 8 rows * 512 f32 = 16 KB = 1024 lanes x 16 B.
  const uint32_t hbase = (uint32_t)(uintptr_t)hist;
  const uint32_t ldsA = (uint32_t)(uintptr_t)stg0 + (uint32_t)tid * 16u;
  const uint32_t ldsB = (uint32_t)(uintptr_t)stg1 + (uint32_t)tid * 16u;

  // ---- prime the pipeline: chunk 0 -> buffer 0 ----
  {
    uint64_t ga = (uint64_t)(uintptr_t)src + (uint64_t)tid * 16u;
    asm volatile("global_load_async_to_lds_b128 %0, %1, off"
                 :: "v"(ldsA), "v"(ga) : "memory");
  }

  for (int j = 0; j < NCHUNK; ++j) {
    if (j + 1 < NCHUNK) {
      // issue next chunk into the other buffer, then wait for current chunk
      uint64_t ga = (uint64_t)(uintptr_t)(src + (size_t)(j + 1) * (CH_ROWS * IMG_W))
                  + (uint64_t)tid * 16u;
      uint32_t dst = (j & 1) ? ldsA : ldsB;
      asm volatile("global_load_async_to_lds_b128 %0, %1, off"
                   :: "v"(dst), "v"(ga) : "memory");
      asm volatile("s_wait_asynccnt 1" ::: "memory");  // chunk j done (in-order)
    } else {
      asm volatile("s_wait_asynccnt 0" ::: "memory");  // last chunk done
    }
    __syncthreads();  // chunk j visible from every wave's async engine

    const float4* a4 = (const float4*)((j & 1) ? stg1 : stg0);
    const float4* b4 = a4 + IMG_W / 4;   // one row below
    // 896 float4 pair-groups per chunk; lanes 896..1023 idle this phase
    // (DS atomic throughput, not issue slots, is the limiter).
    for (int g = tid; g < (CH_ROWS * IMG_W) / 4; g += TPB) {
      float4 va = a4[g];
      float4 vb = b4[g];
      hist_accum(hbase, va.x, vb.x);
      hist_accum(hbase, va.y, vb.y);
      hist_accum(hbase, va.z, vb.z);
      hist_accum(hbase, va.w, vb.w);
    }
    __syncthreads();  // all reads of this buffer done before it is re-targeted
  }

  // Drain our asm-issued ds_add_f32 ops (compiler can't track them), then
  // make all waves' updates visible before the scan.
  asm volatile("s_wait_dscnt 0" ::: "memory");
  __syncthreads();

  // ---- max over the 256x256 histogram (ds_load_b128) ----
  float m = 0.0f;     // histogram entries are >= 0
  #pragma unroll
  for (int k = 0; k < HIST_ELEMS / (4 * TPB); ++k) {
    float4 v = h4[k * TPB + tid];
    m = fmaxf(m, fmaxf(fmaxf(v.x, v.y), fmaxf(v.z, v.w)));
  }

  float* red = stg0;  // staging buffers are free now
  red[tid] = m;
  __syncthreads();
  for (int s = TPB >> 1; s > 0; s >>= 1) {
    if (tid < s) red[tid] = fmaxf(red[tid], red[tid + s]);
    __syncthreads();
  }
  const float inv = 1.0f / red[0];

  // ---- normalized, coalesced b128 store ----
  float4* o4 = (float4*)(out + (size_t)img * HIST_ELEMS);
  #pragma unroll
  for (int k = 0; k < HIST_ELEMS / (4 * TPB); ++k) {
    float4 v = h4[k * TPB + tid];
    v.x *= inv; v.y *= inv; v.z *= inv; v.w *= inv;
    o4[k * TPB + tid] = v;
  }
}

extern "C" void kernel_launch(void* const* d_in, const int* in_sizes, int n_in,
                              void* d_out, int out_size, void* d_ws, size_t ws_size,
                              hipStream_t stream) {
  (void)n_in; (void)out_size; (void)d_ws; (void)ws_size;
  const float* X = (const float*)d_in[0];
  float* out = (float*)d_out;
  const int nimg = in_sizes[0] / (IMG_H * IMG_W);   // B*C = 96
  cooccur_hist_kernel<<<dim3(nimg), dim3(TPB), LDS_BYTES, stream>>>(X, out);
}